// NDPushPullLoss_1022202216835
// MI455X (gfx1250) — compile-verified
//
#include <hip/hip_runtime.h>
#include <hip/hip_bf16.h>

// ---- problem constants -------------------------------------------------
#define BB   8                    // batch
#define NCH  16                   // feature channels
#define HH   368
#define WW   640
#define PP   (HH * WW)            // 235520 pixels, divisible by 256 and 8
#define CIN  12                   // instance ids 1..12
#define CPAD 16                   // padded instance dim (WMMA column dim)

#define MARGIN_VAR  0.5f
#define MARGIN_DIST 3.0f

// ws layout (floats)
#define WS_SUMS    0              // [B][16][16]  = 2048
#define WS_COUNTS  2048           // [B][16]      = 128
#define WS_CENTERS 2176           // [B][16][16]  = 2048
#define WS_CENSQ   4224           // [B][16]      = 128
#define WS_HINGE   4352           // [B][16]      = 128
#define WS_PUSH    4480           // {num, npairs}= 2
#define WS_TOTAL   4482

typedef __attribute__((ext_vector_type(2))) float v2f;
typedef __attribute__((ext_vector_type(8))) float v8f;

// ---- K0: zero scratch ---------------------------------------------------
__global__ void k_zero(float* __restrict__ ws, int n) {
    int i = blockIdx.x * blockDim.x + threadIdx.x;
    if (i < n) ws[i] = 0.0f;
}

// ---- K1: per-instance feature sums + counts via f32 WMMA ----------------
// D[chan, inst] += A(16x4 feat) x B(4x16 mask), accumulated over all pixels.
// The K->pixel bijection inside each WMMA is chosen so every lane reads one
// contiguous float4 of feat and one int4 of labels per 8-pixel step, feeding
// TWO independent WMMA accumulators:
//   WMMA#0: half0 K={0,1}->pix p0+0,1 ; half1 K={2,3}->pix p0+4,5
//   WMMA#1: half0 K={0,1}->pix p0+2,3 ; half1 K={2,3}->pix p0+6,7
#define BLKS_PER_B 64
#define WAVES_PER_B (BLKS_PER_B * 8)

__global__ void k_centersum(const float* __restrict__ feat,
                            const int*   __restrict__ gt,
                            float* __restrict__ sums,
                            float* __restrict__ counts) {
    const int b    = blockIdx.x / BLKS_PER_B;
    const int bb   = blockIdx.x % BLKS_PER_B;
    const int wave = threadIdx.x >> 5;
    const int lane = threadIdx.x & 31;
    const int half = lane >> 4;     // K pair selector
    const int m    = lane & 15;     // A row (channel) / B col (instance)

    const float* fb = feat + (size_t)b * NCH * PP;
    const int*   gb = gt   + (size_t)b * PP;

    const int wgid    = bb * 8 + wave;
    const int ngroups = PP / 8;     // 8 pixels per iteration

    v8f   acc0 = {};
    v8f   acc1 = {};
    float cnt  = 0.0f;

    for (int g = wgid; g < ngroups; g += WAVES_PER_B) {
        const int p0 = g * 8 + half * 4;

        // one coalesced 16B feat load and one 16B label load per lane
        const float4 f4 = *(const float4*)(fb + (size_t)m * NCH * 0 + (size_t)m * PP + p0);
        const int4   l4 = *(const int4*)(gb + p0);

        v2f a0; a0.x = f4.x; a0.y = f4.y;
        v2f a1; a1.x = f4.z; a1.y = f4.w;

        v2f b0, b1;
        b0.x = (l4.x == m + 1) ? 1.0f : 0.0f;
        b0.y = (l4.y == m + 1) ? 1.0f : 0.0f;
        b1.x = (l4.z == m + 1) ? 1.0f : 0.0f;
        b1.y = (l4.w == m + 1) ? 1.0f : 0.0f;
        cnt += (b0.x + b0.y) + (b1.x + b1.y);

        acc0 = __builtin_amdgcn_wmma_f32_16x16x4_f32(
                   false, a0, false, b0, (short)0, acc0, false, false);
        acc1 = __builtin_amdgcn_wmma_f32_16x16x4_f32(
                   false, a1, false, b1, (short)0, acc1, false, false);
    }

    const v8f acc = acc0 + acc1;

    // Scatter: VGPR r, lanes 0-15 -> chan M=r ; lanes 16-31 -> chan M=r+8.
    float* sdst = sums + ((size_t)b * CPAD + m) * NCH;
#pragma unroll
    for (int r = 0; r < 8; ++r)
        atomicAdd(&sdst[r + 8 * half], acc[r]);

    atomicAdd(&counts[b * CPAD + m], cnt);   // halves cover disjoint pixels
}

// ---- K2: centers, center norms, push loss (single block) ----------------
__global__ void k_centers_push(const float* __restrict__ sums,
                               const float* __restrict__ counts,
                               float* __restrict__ centers,
                               float* __restrict__ censq,
                               float* __restrict__ pushv) {
    __shared__ float sc[BB * CPAD * NCH];   // 2048 floats
    __shared__ float rs[256];
    __shared__ float rp[256];
    const int tid = threadIdx.x;

    for (int i = tid; i < BB * CPAD * NCH; i += 256) {
        const int bc = i >> 4;              // b*16 + c
        const float c = counts[bc];
        const float v = sums[i] / fmaxf(c, 1.0f);
        sc[i] = v;
        centers[i] = v;
    }
    __syncthreads();

    for (int i = tid; i < BB * CPAD; i += 256) {
        float s = 0.0f;
#pragma unroll
        for (int n = 0; n < NCH; ++n) { float v = sc[i * NCH + n]; s += v * v; }
        censq[i] = s;
    }
    __syncthreads();

    float psum = 0.0f, pcnt = 0.0f;
    for (int idx = tid; idx < BB * CIN * CIN; idx += 256) {
        const int b = idx / (CIN * CIN);
        const int r = idx % (CIN * CIN);
        const int i = r / CIN, j = r % CIN;
        if (i != j && counts[b * CPAD + i] > 0.0f && counts[b * CPAD + j] > 0.0f) {
            float sq = 0.0f;
            const float* ci = &sc[(b * CPAD + i) * NCH];
            const float* cj = &sc[(b * CPAD + j) * NCH];
#pragma unroll
            for (int n = 0; n < NCH; ++n) { float d = ci[n] - cj[n]; sq += d * d; }
            const float pd = (sq > 0.0f) ? sqrtf(sq) : 0.0f;
            psum += fmaxf(2.0f * MARGIN_DIST - pd, 0.0f);
            pcnt += 1.0f;
        }
    }
    rs[tid] = psum; rp[tid] = pcnt;
    __syncthreads();
    for (int s = 128; s > 0; s >>= 1) {
        if (tid < s) { rs[tid] += rs[tid + s]; rp[tid] += rp[tid + s]; }
        __syncthreads();
    }
    if (tid == 0) { pushv[0] = rs[0]; pushv[1] = rp[0]; }
}

// ---- K3: pull pass (one thread per pixel; BW-bound, mostly L2 hits) -----
__global__ void k_pull(const float* __restrict__ feat,
                       const int*   __restrict__ gt,
                       const float* __restrict__ centers,
                       const float* __restrict__ censq,
                       float* __restrict__ hinge) {
    const int tid = threadIdx.x;
    const int idx = blockIdx.x * 256 + tid;     // < B*P, P % 256 == 0
    const int b   = idx / PP;
    const int p   = idx % PP;

    __shared__ float ctr[CIN * NCH];
    __shared__ float csq[CIN];
    __shared__ float bins[CIN];

    if (tid < CIN * NCH) ctr[tid] = centers[b * CPAD * NCH + tid];
    if (tid < CIN) { csq[tid] = censq[b * CPAD + tid]; bins[tid] = 0.0f; }
    __syncthreads();

    const int lab = gt[idx];
    if (lab >= 1 && lab <= CIN) {
        const int c = lab - 1;
        const float* f = feat + (size_t)b * NCH * PP + p;
        float dot = 0.0f, fsq = 0.0f;
#pragma unroll
        for (int n = 0; n < NCH; ++n) {
            const float v = f[(size_t)n * PP];   // coalesced across the block
            dot += v * ctr[c * NCH + n];
            fsq += v * v;
        }
        float sq = fmaxf(fsq + csq[c] - 2.0f * dot, 0.0f);
        const float d = (sq > 0.0f) ? sqrtf(sq) : 0.0f;
        const float h = fmaxf(d - MARGIN_VAR, 0.0f);
        atomicAdd(&bins[c], h);                  // LDS atomic
    }
    __syncthreads();
    if (tid < CIN) {
        const float v = bins[tid];
        if (v != 0.0f) atomicAdd(&hinge[b * CPAD + tid], v);
    }
}

// ---- K4: combine --------------------------------------------------------
__global__ void k_final(const float* __restrict__ counts,
                        const float* __restrict__ hinge,
                        const float* __restrict__ pushv,
                        float* __restrict__ out) {
    if (threadIdx.x == 0 && blockIdx.x == 0) {
        float pullnum = 0.0f, ninst = 0.0f;
        for (int b = 0; b < BB; ++b)
            for (int c = 0; c < CIN; ++c) {
                const float cnt = counts[b * CPAD + c];
                if (cnt > 0.0f) { ninst += 1.0f; pullnum += hinge[b * CPAD + c] / cnt; }
            }
        const float pull = (ninst > 0.0f) ? pullnum / ninst : 0.0f;
        const float push = (pushv[1] > 0.0f) ? pushv[0] / pushv[1] : 0.0f;
        out[0] = push + pull;                   // DIST_WEIGHT = VAR_WEIGHT = 1
    }
}

// ---- launch -------------------------------------------------------------
extern "C" void kernel_launch(void* const* d_in, const int* in_sizes, int n_in,
                              void* d_out, int out_size, void* d_ws, size_t ws_size,
                              hipStream_t stream) {
    const float* feat = (const float*)d_in[0];
    const int*   gt   = (const int*)d_in[1];
    float* ws      = (float*)d_ws;
    float* sums    = ws + WS_SUMS;
    float* counts  = ws + WS_COUNTS;
    float* centers = ws + WS_CENTERS;
    float* censq   = ws + WS_CENSQ;
    float* hinge   = ws + WS_HINGE;
    float* pushv   = ws + WS_PUSH;
    float* out     = (float*)d_out;

    k_zero<<<(WS_TOTAL + 255) / 256, 256, 0, stream>>>(ws, WS_TOTAL);
    k_centersum<<<BB * BLKS_PER_B, 256, 0, stream>>>(feat, gt, sums, counts);
    k_centers_push<<<1, 256, 0, stream>>>(sums, counts, centers, censq, pushv);
    k_pull<<<(BB * PP) / 256, 256, 0, stream>>>(feat, gt, centers, censq, hinge);
    k_final<<<1, 64, 0, stream>>>(counts, hinge, pushv, out);
}